// DehLoss_21801253994970
// MI455X (gfx1250) — compile-verified
//
#include <hip/hip_runtime.h>
#include <hip/hip_bf16.h>
#include <math.h>

// ---------------------------------------------------------------------------
// DehLoss: n=8192 pairwise kernel-density / CDF loss.
//   R[i] = g1[i] + log(y[i]); L[i] = exp(g2[i]-g1[i]); d[i] = delta[i]
//   DR[i,j] = (R[i]-R[j])/h ; K = N(DR) pdf ; P = ndtr(DR)
//   Dk[j] = sum_i d[i]K[i,j] ; LP[j] = sum_i L[i]P[i,j]
//   out = -(S1+S2+S3+S4)   (all O(n) post-processing)
// Compute-bound: 64M pairs * (exp + erfc) >> 160KB of HBM traffic.
// Reductions over i are done with V_WMMA_F32_16X16X4_F32: each wave owns a
// 16-column j-tile; per step it builds the 4x16 B tile (2 pairs/lane) and a
// broadcast-d (resp. broadcast-L) A operand, so the f32 C accumulator carries
// the running per-j sums (all rows identical -> no cross-lane reduction).
// ---------------------------------------------------------------------------

#define NN      8192
#define NCH     4
#define CHUNK   (NN / NCH)        // 2048 i per block, staged in LDS (24KB)
#define WPB     8                 // waves per block (256 threads, wave32)
#define TILE_J  16

typedef __attribute__((ext_vector_type(2))) float v2f;
typedef __attribute__((ext_vector_type(8))) float v8f;

#define INV_SQRT_2PI 0.3989422804014327f
#define INV_SQRT_2   0.70710678118654752f

// ---- kernel 0: precompute R, L, d; zero accumulators and output -----------
__global__ __launch_bounds__(256) void deh_init(
    const float* __restrict__ m_z, const float* __restrict__ y,
    const float* __restrict__ delta,
    float* __restrict__ Rv, float* __restrict__ Lv, float* __restrict__ dv,
    float* __restrict__ Dk, float* __restrict__ LP, float* __restrict__ out,
    int n)
{
    int i = blockIdx.x * 256 + threadIdx.x;
    if (i < n) {
        float g1 = m_z[2 * i];
        float g2 = m_z[2 * i + 1];
        Rv[i] = g1 + logf(y[i]);
        Lv[i] = expf(g2 - g1);
        dv[i] = delta[i];
        Dk[i] = 0.0f;
        LP[i] = 0.0f;
    }
    if (i == 0) out[0] = 0.0f;
}

// ---- kernel 1: pairwise tile kernel with WMMA accumulation ----------------
// block -> (i-chunk, group of 8 j-tiles); wave -> one 16-wide j-tile.
__global__ __launch_bounds__(256) void deh_pair(
    const float* __restrict__ Rv, const float* __restrict__ Lv,
    const float* __restrict__ dv,
    float* __restrict__ Dk, float* __restrict__ LP,
    float inv_h)
{
    __shared__ float sR[CHUNK];
    __shared__ float sL[CHUNK];
    __shared__ float sD[CHUNK];

    const int chunk = blockIdx.x & (NCH - 1);
    const int jgrp  = blockIdx.x >> 2;          // / NCH
    const int ibase = chunk * CHUNK;

    // cooperative stage of this block's i-chunk into LDS
    for (int t = threadIdx.x; t < CHUNK; t += 256) {
        sR[t] = Rv[ibase + t];
        sL[t] = Lv[ibase + t];
        sD[t] = dv[ibase + t];
    }
    __syncthreads();

    const int wave = threadIdx.x >> 5;
    const int lane = threadIdx.x & 31;
    const int j    = (jgrp * WPB + wave) * TILE_J + (lane & 15);
    const int h2   = (lane >> 4) << 1;          // slot base: 0 (lanes 0-15) / 2 (16-31)
    const float Rj = Rv[j];

    v8f cK = {};                                // running d^T K partial (per j)
    v8f cP = {};                                // running L^T P partial (per j)

    #pragma unroll 2
    for (int ib = 0; ib < CHUNK; ib += 4) {
        const int i0 = ib + h2;
        // two (i,j) pairs per lane -> the 4x16 B tiles for K and P
        float dr0 = (sR[i0]     - Rj) * inv_h;
        float dr1 = (sR[i0 + 1] - Rj) * inv_h;
        float k0 = INV_SQRT_2PI * __expf(-0.5f * dr0 * dr0);
        float k1 = INV_SQRT_2PI * __expf(-0.5f * dr1 * dr1);
        float p0 = 0.5f * erfcf(-INV_SQRT_2 * dr0);   // ndtr(dr0)
        float p1 = 0.5f * erfcf(-INV_SQRT_2 * dr1);   // ndtr(dr1)

        v2f b_k = { k0, k1 };
        v2f b_p = { p0, p1 };
        // A operands: d (resp. L) broadcast along M, indexed by the same
        // slot->K convention as B, so D[m][n] = sum_k d[i0+k]*K[i0+k][j] forall m.
        v2f a_d = { sD[i0], sD[i0 + 1] };
        v2f a_l = { sL[i0], sL[i0 + 1] };

        cK = __builtin_amdgcn_wmma_f32_16x16x4_f32(
                 false, a_d, false, b_k, (short)0, cK, false, false);
        cP = __builtin_amdgcn_wmma_f32_16x16x4_f32(
                 false, a_l, false, b_p, (short)0, cP, false, false);
    }

    // C/D layout: VGPR0, lanes 0-15 hold (M=0, N=lane) -> full tile partial.
    if (lane < 16) {
        atomicAdd(&Dk[j], cK[0]);
        atomicAdd(&LP[j], cP[0]);
    }
}

// ---- kernel 2: per-j log terms + reduction into the scalar output ---------
__global__ __launch_bounds__(256) void deh_finish(
    const float* __restrict__ m_z, const float* __restrict__ dv,
    const float* __restrict__ Rv,
    const float* __restrict__ Dk, const float* __restrict__ LP,
    float* __restrict__ out, int n, float h)
{
    __shared__ float red[256];
    const int j = blockIdx.x * 256 + threadIdx.x;

    const float invn  = 1.0f / (float)n;
    const float invnh = invn / h;

    float dj = dv[j];
    float g2 = m_z[2 * j + 1];
    float s  = dj * g2                                   // S1 term
             - dj * Rv[j]                                // S2 term
             + dj * logf(Dk[j] * invnh + 1e-15f)         // S3 term
             - dj * logf(LP[j] * invn  + 1e-15f);        // S4 term

    red[threadIdx.x] = s;
    __syncthreads();
    for (int off = 128; off > 0; off >>= 1) {
        if (threadIdx.x < off) red[threadIdx.x] += red[threadIdx.x + off];
        __syncthreads();
    }
    if (threadIdx.x == 0) atomicAdd(out, -red[0] * invn);
}

// ---------------------------------------------------------------------------
extern "C" void kernel_launch(void* const* d_in, const int* in_sizes, int n_in,
                              void* d_out, int out_size, void* d_ws, size_t ws_size,
                              hipStream_t stream) {
    const float* m_z   = (const float*)d_in[0];   // (n,2)
    const float* y     = (const float*)d_in[1];   // (n,1)
    const float* delta = (const float*)d_in[2];   // (n,1)
    float* out = (float*)d_out;
    float* ws  = (float*)d_ws;

    const int n = in_sizes[0] / 2;                // 8192

    float* Rv = ws;
    float* Lv = ws + (size_t)n;
    float* dv = ws + (size_t)2 * n;
    float* Dk = ws + (size_t)3 * n;
    float* LP = ws + (size_t)4 * n;

    const float h = 1.3f * powf((float)n, -0.2f);

    deh_init<<<(n + 255) / 256, 256, 0, stream>>>(m_z, y, delta,
                                                  Rv, Lv, dv, Dk, LP, out, n);

    const int njb = n / (TILE_J * WPB);           // 64 j-groups
    deh_pair<<<njb * NCH, 256, 0, stream>>>(Rv, Lv, dv, Dk, LP, 1.0f / h);

    deh_finish<<<n / 256, 256, 0, stream>>>(m_z, dv, Rv, Dk, LP, out, n, h);
}